// meanPred_12326556140158
// MI455X (gfx1250) — compile-verified
//
#include <hip/hip_runtime.h>
#include <math.h>

typedef float v8f  __attribute__((ext_vector_type(8)));
typedef float v2f  __attribute__((ext_vector_type(2)));

constexpr int T = 32, P = 1024, Q = 512, D = 1024;
constexpr int SPLIT = 8;
constexpr int ROWS_PER_SPLIT = P / SPLIT;   // 128

// ---------------- zero scratch accumulators ----------------
__global__ void k_zero(float* __restrict__ p, int n) {
    int i = blockIdx.x * blockDim.x + threadIdx.x;
    if (i < n) p[i] = 0.0f;
}

// ---------------- per-row inverse norms of pool ----------------
// one wave (32 lanes) per row; 8 float4 loads per lane, shfl reduce
__global__ void __launch_bounds__(256) k_norm(const float* __restrict__ pool,
                                              float* __restrict__ inv_pn) {
    int wv = threadIdx.x >> 5, lane = threadIdx.x & 31;
    int row = blockIdx.x * 8 + wv;                      // grid = T*P/8
    const float4* rp = (const float4*)(pool + (size_t)row * D);
    float s = 0.0f;
#pragma unroll
    for (int i = 0; i < 8; ++i) {
        float4 v = rp[lane + i * 32];
        s = fmaf(v.x, v.x, s); s = fmaf(v.y, v.y, s);
        s = fmaf(v.z, v.z, s); s = fmaf(v.w, v.w, s);
    }
#pragma unroll
    for (int off = 16; off; off >>= 1) s += __shfl_down(s, off, 32);
    if (lane == 0) inv_pn[row] = 1.0f / sqrtf(s);
}

// ---------------- per-task pos/neg counts ----------------
__global__ void k_count(const int* __restrict__ mask, const int* __restrict__ plab,
                        float* __restrict__ counts) {
    __shared__ int sp[256], sn[256];
    int t = blockIdx.x, tid = threadIdx.x;
    int cp = 0, cn = 0;
    for (int p = tid; p < P; p += 256) {
        int m = mask[t * P + p] != 0;
        int l = plab[t * P + p] != 0;
        cp += m & l;
        cn += m & (l ^ 1);
    }
    sp[tid] = cp; sn[tid] = cn; __syncthreads();
    for (int s = 128; s > 0; s >>= 1) {
        if (tid < s) { sp[tid] += sp[tid + s]; sn[tid] += sn[tid + s]; }
        __syncthreads();
    }
    if (tid == 0) { counts[t * 2] = (float)sp[0]; counts[t * 2 + 1] = (float)sn[0]; }
}

// ---------------- masked normalized column sums ----------------
// thread owns column d of task t; loops 128 rows of its P-split; fully coalesced
__global__ void __launch_bounds__(256) k_accum(const float* __restrict__ pool,
                                               const int* __restrict__ mask,
                                               const int* __restrict__ plab,
                                               const float* __restrict__ inv_pn,
                                               float* __restrict__ wpos,
                                               float* __restrict__ wneg) {
    int t  = blockIdx.x;
    int d  = blockIdx.y * 256 + threadIdx.x;
    int p0 = blockIdx.z * ROWS_PER_SPLIT;
    const float* base = pool + (size_t)t * P * D;
    float ap = 0.0f, an = 0.0f;
    for (int p = p0; p < p0 + ROWS_PER_SPLIT; ++p) {
        int   m  = mask[t * P + p];        // uniform across block -> scalar loads
        int   l  = plab[t * P + p];
        float ip = inv_pn[t * P + p];
        float v  = base[(size_t)p * D + d];
        float sp = (m != 0 && l != 0) ? ip : 0.0f;
        float sn = (m != 0 && l == 0) ? ip : 0.0f;
        ap = fmaf(v, sp, ap);
        an = fmaf(v, sn, an);
    }
    atomicAdd(&wpos[t * D + d], ap);
    atomicAdd(&wneg[t * D + d], an);
}

// ---------------- combine into single weight vector ----------------
__global__ void k_finalw(const float* __restrict__ wpos, const float* __restrict__ wneg,
                         const float* __restrict__ counts, float* __restrict__ w) {
    int t = blockIdx.x, d = blockIdx.y * 256 + threadIdx.x;
    float np = counts[t * 2], nn = counts[t * 2 + 1];
    float ip = np > 0.0f ? 1.0f / np : 0.0f;   // 0/0 -> NaN -> 0 rule
    float in_ = nn > 0.0f ? 1.0f / nn : 0.0f;
    w[t * D + d] = wpos[t * D + d] * ip - wneg[t * D + d] * in_;
}

// ---------------- WMMA scoring: score = (q . w) / ||q|| ----------------
// Wave handles a 16-query tile. A = 16x4 f32 query slab (ISA layout:
// M = lane%16, K-pair = 2*(lane/16)); B column 0 = w chunk, columns 1..15
// read a zeroed LDS pad via branchless index select (EXEC stays all-1s).
__global__ void __launch_bounds__(256) k_score(const float* __restrict__ query,
                                               const float* __restrict__ w,
                                               float* __restrict__ out) {
    __shared__ __align__(16) float wl[D + 8];   // [D..D+1] = zeros for dead columns
    __shared__ float dots[8][16];
    int t = blockIdx.x;
    int tid = threadIdx.x, wv = tid >> 5, lane = tid & 31;

    for (int i = tid; i < D; i += 256) wl[i] = w[t * D + i];
    if (tid < 8) wl[D + tid] = 0.0f;
    __syncthreads();

    int tile  = blockIdx.y * 8 + wv;            // 32 tiles/task
    int m     = lane & 15;                      // A row / B column for this lane
    int khalf = (lane >> 4) * 2;                // K 0..1 vs 2..3
    const float* qrow = query + ((size_t)t * Q + (size_t)tile * 16 + m) * D;
    int nmask = (m == 0) ? ~0 : 0;              // lane carries real B only in col 0

    v8f  c   = {};
    float ssq = 0.0f;
    for (int kb = 0; kb < D; kb += 4) {
        int ka = kb + khalf;
        v2f a = *(const v2f*)(qrow + ka);
        v2f b = *(const v2f*)(wl + ((ka & nmask) | (D & ~nmask)));
        ssq = fmaf(a.x, a.x, ssq);
        ssq = fmaf(a.y, a.y, ssq);
        c = __builtin_amdgcn_wmma_f32_16x16x4_f32(false, a, false, b,
                                                  (short)0, c, false, false);
    }

    // total ||q||^2 for row m lives split across lanes m and m+16
    ssq += __shfl_xor(ssq, 16, 32);

    // column 0 of D: lane 0 holds rows 0..7, lane 16 holds rows 8..15
    if (lane == 0) {
#pragma unroll
        for (int i = 0; i < 8; ++i) dots[wv][i] = c[i];
    } else if (lane == 16) {
#pragma unroll
        for (int i = 0; i < 8; ++i) dots[wv][8 + i] = c[i];
    }
    __syncthreads();

    if (lane < 16)
        out[(size_t)t * Q + (size_t)tile * 16 + lane] = dots[wv][lane] * rsqrtf(ssq);
}

extern "C" void kernel_launch(void* const* d_in, const int* in_sizes, int n_in,
                              void* d_out, int out_size, void* d_ws, size_t ws_size,
                              hipStream_t stream) {
    (void)in_sizes; (void)n_in; (void)out_size; (void)ws_size;
    const float* pool  = (const float*)d_in[0];
    const float* query = (const float*)d_in[1];
    const int*   mask  = (const int*)d_in[2];
    const int*   plab  = (const int*)d_in[3];
    // d_in[4] (query_labels) is unused by the reference
    float* out = (float*)d_out;

    float* ws     = (float*)d_ws;
    float* inv_pn = ws;                    // T*P      = 32768
    float* wpos   = inv_pn + T * P;        // T*D      = 32768
    float* wneg   = wpos   + T * D;        // T*D      = 32768
    float* counts = wneg   + T * D;        // T*2      = 64
    float* wvec   = counts + 2 * T;        // T*D      = 32768

    k_zero  <<<(2 * T * D + 255) / 256, 256, 0, stream>>>(wpos, 2 * T * D);
    k_norm  <<<T * P / 8, 256, 0, stream>>>(pool, inv_pn);
    k_count <<<T, 256, 0, stream>>>(mask, plab, counts);
    k_accum <<<dim3(T, D / 256, SPLIT), 256, 0, stream>>>(pool, mask, plab,
                                                          inv_pn, wpos, wneg);
    k_finalw<<<dim3(T, D / 256), 256, 0, stream>>>(wpos, wneg, counts, wvec);
    k_score <<<dim3(T, Q / 128), 256, 0, stream>>>(query, wvec, out);
}